// GCN_35820027248930
// MI455X (gfx1250) — compile-verified
//
#include <hip/hip_runtime.h>
#include <hip/hip_bf16.h>

typedef __attribute__((ext_vector_type(2))) float v2f;
typedef __attribute__((ext_vector_type(8))) float v8f;

// ---------------- gcn_norm kernels ----------------

__global__ void k_fill1(float* __restrict__ deg, int n) {
  int i = blockIdx.x * blockDim.x + threadIdx.x;
  if (i < n) deg[i] = 1.0f;  // self-loop weight
}

__global__ void k_degacc(const long long* __restrict__ col, const float* __restrict__ w,
                         float* __restrict__ deg, int E) {
  int e = blockIdx.x * blockDim.x + threadIdx.x;
  if (e < E) atomicAdd(&deg[(size_t)col[e]], w[e]);
}

__global__ void k_dinv(const float* __restrict__ deg, float* __restrict__ dinv, int n) {
  int i = blockIdx.x * blockDim.x + threadIdx.x;
  if (i < n) {
    float d = deg[i];
    dinv[i] = d > 0.f ? rsqrtf(d) : 0.f;
  }
}

__global__ void k_norm(const long long* __restrict__ row, const long long* __restrict__ col,
                       const float* __restrict__ w, const float* __restrict__ dinv,
                       float* __restrict__ nrm, int E) {
  int e = blockIdx.x * blockDim.x + threadIdx.x;
  if (e < E) nrm[e] = dinv[(size_t)row[e]] * w[e] * dinv[(size_t)col[e]];
}

// ---------------- WMMA GEMM: H[M,NC] = X[M,K] @ W[K,NC] ----------------
// One wave -> 16-row strip, all NC columns (NC/16 accumulator tiles).
// f32 WMMA 16x16x4. W staged through LDS in KC-row chunks, K-PAIR INTERLEAVED:
//   sW[(kp*NC + n)*2 + 0] = W[k0+2kp][n]
//   sW[(kp*NC + n)*2 + 1] = W[k0+2kp+1][n]
// so a B fragment {W[k][n], W[k+1][n]} is one aligned ds_load_b64 (no shuffle movs).

template <int K, int NC, int KC>
__global__ void k_gemm_wmma(const float* __restrict__ X, const float* __restrict__ Wt,
                            float* __restrict__ H, int M) {
  constexpr int NT = NC / 16;
  __shared__ float sW[KC * NC];

  const int tid = threadIdx.x;
  const int lane = tid & 31;
  const int wave = blockIdx.x * (blockDim.x >> 5) + (tid >> 5);
  const int m0 = wave * 16;
  const bool active = (m0 < M);          // wave-uniform -> EXEC all-ones inside
  const int m = lane & 15;
  const int kh = lane >> 4;              // half-wave selects K pair {0,1} vs {2,3}

  v8f acc[NT] = {};

  const float* xp = X + (size_t)(active ? (m0 + m) : 0) * K + 2 * kh;

  for (int k0 = 0; k0 < K; k0 += KC) {
    __syncthreads();
    // cooperative fill with k-pair interleave: each item covers 2 rows x 4 cols
    constexpr int ITEMS = (KC / 2) * (NC / 4);
    for (int idx = tid; idx < ITEMS; idx += blockDim.x) {
      int kp = idx / (NC / 4);
      int n  = (idx % (NC / 4)) * 4;
      float4 a = *(const float4*)(Wt + (size_t)(k0 + 2 * kp + 0) * NC + n);
      float4 b = *(const float4*)(Wt + (size_t)(k0 + 2 * kp + 1) * NC + n);
      float4 lo = make_float4(a.x, b.x, a.y, b.y);
      float4 hi = make_float4(a.z, b.z, a.w, b.w);
      *(float4*)(sW + (size_t)(kp * NC + n) * 2 + 0) = lo;
      *(float4*)(sW + (size_t)(kp * NC + n) * 2 + 4) = hi;
    }
    __syncthreads();

    if (active) {
#pragma unroll 4
      for (int k = 0; k < KC; k += 4) {
        float2 av = *(const float2*)(xp + k0 + k);   // A frag: K{0,1} or K{2,3}
        v2f a; a[0] = av.x; a[1] = av.y;
        const int kp = (k >> 1) + kh;                // interleaved pair row
#pragma unroll
        for (int t = 0; t < NT; ++t) {
          v2f b = *(const v2f*)(sW + (size_t)(kp * NC + 16 * t + m) * 2);
          acc[t] = __builtin_amdgcn_wmma_f32_16x16x4_f32(
              false, a, false, b, (short)0, acc[t], false, false);
        }
      }
    }
  }

  if (active) {
    const int rb = kh * 8;  // C/D layout: lanes16-31 hold M=8..15
#pragma unroll
    for (int t = 0; t < NT; ++t)
#pragma unroll
      for (int i = 0; i < 8; ++i)
        H[(size_t)(m0 + rb + i) * NC + 16 * t + m] = acc[t][i];
  }
}

// ---------------- aggregation ----------------

// out[i,c] = h[i,c]*dinv[i]^2 (+ bias[c])  : self-loop message + optional bias
template <int C>
__global__ void k_selfloop(const float* __restrict__ h, const float* __restrict__ dinv,
                           const float* __restrict__ bias, float* __restrict__ out, int total) {
  int idx = blockIdx.x * blockDim.x + threadIdx.x;
  if (idx >= total) return;
  int node = idx >> (C == 128 ? 7 : 6);
  int c = idx & (C - 1);
  float d = dinv[node];
  float v = h[idx] * d * d;
  if (bias) v += bias[c];
  out[idx] = v;
}

// one wave per edge; each lane covers C/32 channels (float4 / float2 gather, f32 atomics)
template <int C>
__global__ void k_agg_edges(const float* __restrict__ h, const long long* __restrict__ row,
                            const long long* __restrict__ col, const float* __restrict__ nrm,
                            float* __restrict__ out, int E) {
  int t = blockIdx.x * blockDim.x + threadIdx.x;
  int e = t >> 5;
  if (e >= E) return;
  int lane = t & 31;
  constexpr int V = C / 32;
  size_t r = (size_t)row[e], c = (size_t)col[e];
  float wn = nrm[e];
  const float* hp = h + r * C + lane * V;
  float* op = out + c * C + lane * V;
  if constexpr (V == 4) {
    float4 v = *(const float4*)hp;
    atomicAdd(op + 0, v.x * wn);
    atomicAdd(op + 1, v.y * wn);
    atomicAdd(op + 2, v.z * wn);
    atomicAdd(op + 3, v.w * wn);
  } else {
    float2 v = *(const float2*)hp;
    atomicAdd(op + 0, v.x * wn);
    atomicAdd(op + 1, v.y * wn);
  }
}

template <int C>
__global__ void k_bias_relu(float* __restrict__ a, const float* __restrict__ bias, int total) {
  int idx = blockIdx.x * blockDim.x + threadIdx.x;
  if (idx >= total) return;
  float v = a[idx] + bias[idx & (C - 1)];
  a[idx] = v > 0.f ? v : 0.f;
}

// ---------------- launch ----------------

extern "C" void kernel_launch(void* const* d_in, const int* in_sizes, int n_in,
                              void* d_out, int out_size, void* d_ws, size_t ws_size,
                              hipStream_t stream) {
  const float* x        = (const float*)d_in[0];
  const long long* ei   = (const long long*)d_in[1];   // int64 [2,E]
  const float* ew       = (const float*)d_in[2];
  const float* W1       = (const float*)d_in[3];
  const float* b1       = (const float*)d_in[4];
  const float* W2       = (const float*)d_in[5];
  const float* b2       = (const float*)d_in[6];
  float* out            = (float*)d_out;

  const int IN = 256, HID = 128, OUT = 64;
  const int N = in_sizes[0] / IN;   // 50000
  const int E = in_sizes[2];        // 800000
  const long long* row = ei;
  const long long* colp = ei + E;

  auto al4 = [](size_t v) { return (v + 3) & ~(size_t)3; };
  float* ws = (float*)d_ws;
  size_t o = 0;
  float* deg  = ws + o; o = al4(o + N);
  float* dinv = ws + o; o = al4(o + N);
  float* nrm  = ws + o; o = al4(o + E);
  float* h1   = ws + o; o += (size_t)N * HID;
  float* a1   = ws + o; o += (size_t)N * HID;
  float* h2   = ws + o; o += (size_t)N * OUT;
  (void)ws_size;

  const int B = 256;
  dim3 blk(B);

  // gcn_norm
  k_fill1<<<dim3((N + B - 1) / B), blk, 0, stream>>>(deg, N);
  k_degacc<<<dim3((E + B - 1) / B), blk, 0, stream>>>(colp, ew, deg, E);
  k_dinv<<<dim3((N + B - 1) / B), blk, 0, stream>>>(deg, dinv, N);
  k_norm<<<dim3((E + B - 1) / B), blk, 0, stream>>>(row, colp, ew, dinv, nrm, E);

  // layer 1: h1 = x @ W1  (WMMA), aggregate into a1, bias+relu
  {
    int waves = (N + 15) / 16;                       // 3125
    int blocks = (waves + 3) / 4;                    // 4 waves / 128-thread block
    k_gemm_wmma<256, 128, 64><<<dim3(blocks), dim3(128), 0, stream>>>(x, W1, h1, N);
  }
  k_selfloop<128><<<dim3(((size_t)N * HID + B - 1) / B), blk, 0, stream>>>(h1, dinv, nullptr, a1, N * HID);
  k_agg_edges<128><<<dim3(((size_t)E * 32 + B - 1) / B), blk, 0, stream>>>(h1, row, colp, nrm, a1, E);
  k_bias_relu<128><<<dim3(((size_t)N * HID + B - 1) / B), blk, 0, stream>>>(a1, b1, N * HID);

  // layer 2: h2 = a1 @ W2 (WMMA), aggregate (+bias) directly into d_out
  {
    int waves = (N + 15) / 16;
    int blocks = (waves + 3) / 4;
    k_gemm_wmma<128, 64, 64><<<dim3(blocks), dim3(128), 0, stream>>>(a1, W2, h2, N);
  }
  k_selfloop<64><<<dim3(((size_t)N * OUT + B - 1) / B), blk, 0, stream>>>(h2, dinv, b2, out, N * OUT);
  k_agg_edges<64><<<dim3(((size_t)E * 32 + B - 1) / B), blk, 0, stream>>>(h2, row, colp, nrm, out, E);
}